// WeakREShead_31559419691040
// MI455X (gfx1250) — compile-verified
//
#include <hip/hip_runtime.h>
#include <math.h>

typedef float v2f __attribute__((ext_vector_type(2)));
typedef float v8f __attribute__((ext_vector_type(8)));

// D = A(16x4 f32) * B(4x16 f32) + C, full f32 precision (V_WMMA_F32_16X16X4_F32)
__device__ __forceinline__ v8f wmma_k4(v2f a, v2f b, v8f c) {
  return __builtin_amdgcn_wmma_f32_16x16x4_f32(
      /*neg_a=*/false, a, /*neg_b=*/false, b,
      /*c_mod=*/(short)0, c, /*reuse_a=*/false, /*reuse_b=*/false);
}

// ---------------------------------------------------------------------------
// Kernel 1: sim[b, a, v] = dot(vis[a, v, :], lan[b, :])
// GEMM: M = a*64+v (2048), N = b (32), K = 256. One wave per 16x16 tile.
// sim stored as sim[b*2048 + a*64 + v].
// ---------------------------------------------------------------------------
__global__ __launch_bounds__(32) void k_sim(const float* __restrict__ vis,
                                            const float* __restrict__ lan,
                                            float* __restrict__ sim) {
  const int mt   = blockIdx.x;        // 0..127
  const int nt   = blockIdx.y;        // 0..1
  const int lane = threadIdx.x;       // 0..31
  const int half = lane >> 4;
  const int l15  = lane & 15;
  const int arow = mt * 16 + l15;     // M row for A fragment
  const int bcol = nt * 16 + l15;     // N col for B fragment

  v8f acc = {0.f, 0.f, 0.f, 0.f, 0.f, 0.f, 0.f, 0.f};
  for (int k0 = 0; k0 < 256; k0 += 4) {
    const int kk = k0 + half * 2;
    v2f a = *(const v2f*)(vis + (size_t)arow * 256 + kk);
    v2f b = *(const v2f*)(lan + (size_t)bcol * 256 + kk);
    acc = wmma_k4(a, b, acc);
  }
  // D layout: VGPR j -> M = mt*16 + j + 8*half, N = bcol
  const int mbase = mt * 16 + 8 * half;
#pragma unroll
  for (int j = 0; j < 8; ++j) {
    sim[(size_t)bcol * 2048 + mbase + j] = acc[j];
  }
}

// ---------------------------------------------------------------------------
// Kernel 2: stable descending sort of each 64-long sim row (matches top_k).
// One block (64 threads) per (b, a) pair.
// ---------------------------------------------------------------------------
__global__ __launch_bounds__(64) void k_sort(const float* __restrict__ sim,
                                             float* __restrict__ lsort,
                                             int* __restrict__ sidx) {
  __shared__ float vals[64];
  const int pair = blockIdx.x;   // b*32 + a
  const int t    = threadIdx.x;
  const float v  = sim[(size_t)pair * 64 + t];
  vals[t] = v;
  __syncthreads();
  int rank = 0;
  for (int j = 0; j < 64; ++j) {
    float vj = vals[j];
    rank += (vj > v) || (vj == v && j < t);
  }
  lsort[(size_t)pair * 64 + rank] = v;
  sidx[(size_t)pair * 64 + rank] = t;
}

// ---------------------------------------------------------------------------
// Kernel 3: l2-normalize each vis row (2048 rows x 256). One block per row.
// ---------------------------------------------------------------------------
__global__ __launch_bounds__(256) void k_norm(const float* __restrict__ vis,
                                              float* __restrict__ vnorm) {
  __shared__ float red[256];
  const int row = blockIdx.x;
  const int t   = threadIdx.x;
  const float x = vis[(size_t)row * 256 + t];
  red[t] = x * x;
  __syncthreads();
  for (int s = 128; s > 0; s >>= 1) {
    if (t < s) red[t] += red[t + s];
    __syncthreads();
  }
  const float n = fmaxf(sqrtf(red[0]), 1e-12f);
  vnorm[(size_t)row * 256 + t] = x / n;
}

// ---------------------------------------------------------------------------
// Kernel 4: greedy hard-negative selection + per-batch loss.
// One block (256 threads = 8 waves) per batch b.
// ---------------------------------------------------------------------------
__global__ __launch_bounds__(256) void k_sel(const float* __restrict__ vnorm,
                                             const float* __restrict__ lsort,
                                             const int* __restrict__ sidx,
                                             float* __restrict__ lossb) {
  __shared__ float diff[1984];      // difficulty[n*64+v]
  __shared__ float uniq[1984];      // uniqueness[n*64+v]
  __shared__ int   sidx_l[1984];    // sorted index per (n, v)
  __shared__ float Ys[32 * 256];    // selected normalized rows (row 31 zero pad)
  __shared__ int   sel_all[4 * 31]; // selections per iteration

  const int b    = blockIdx.x;
  const int t    = threadIdx.x;
  const int lane = t & 31;
  const int w    = t >> 5;          // wave id 0..7
  const int half = lane >> 4;
  const int l15  = lane & 15;

  // Phase 0: difficulty, uniqueness=1, local sorted indices.
  for (int i = t; i < 1984; i += 256) {
    const int n = i >> 6, v = i & 63;
    const int a = n + (n >= b);
    const int base = (b * 32 + a) * 64;
    const float mx = lsort[base];
    const float mn = lsort[base + 63];
    diff[i]   = (lsort[base + v] - mn) / (mx - mn);
    uniq[i]   = 1.0f;
    sidx_l[i] = sidx[base + v];
  }
  Ys[31 * 256 + t] = 0.0f;   // zero the pad row (256 threads cover 256 floats)
  __syncthreads();

  for (int it = 0; it < 4; ++it) {
    // Phase 1: per-n argmax of uniq*diff (first index wins ties, as jnp.argmax).
    if (t < 31) {
      float best = -1e30f;
      int bi = 0;
      for (int v = 0; v < 64; ++v) {
        const float s = uniq[t * 64 + v] * diff[t * 64 + v];
        if (s > best) { best = s; bi = v; }
      }
      sel_all[it * 31 + t] = bi;
    }
    __syncthreads();
    if (it == 3) break;   // last iteration only records selections

    // Phase 2: gather 31 selected normalized rows into LDS.
    for (int i = t; i < 31 * 256; i += 256) {
      const int n2 = i >> 8, d = i & 255;
      const int a2 = n2 + (n2 >= b);
      const int g  = a2 * 64 + sidx_l[n2 * 64 + sel_all[it * 31 + n2]];
      Ys[i] = vnorm[(size_t)g * 256 + d];
    }
    __syncthreads();

    // Phase 3: X(1984x256) . Ys^T(256x32) via WMMA; uniq = min(uniq, 0.5*(1-max_col)).
    for (int mt = w; mt < 124; mt += 8) {   // wave-uniform loop, EXEC all ones
      const int m = mt * 16 + l15;
      const int n = m >> 6, v = m & 63;
      const int a = n + (n >= b);
      const int g = a * 64 + sidx_l[n * 64 + v];
      const float* xrow = vnorm + (size_t)g * 256;
      v8f c0 = {0.f, 0.f, 0.f, 0.f, 0.f, 0.f, 0.f, 0.f};
      v8f c1 = {0.f, 0.f, 0.f, 0.f, 0.f, 0.f, 0.f, 0.f};
      for (int k0 = 0; k0 < 256; k0 += 4) {
        const int kk = k0 + half * 2;
        v2f av = *(const v2f*)(xrow + kk);
        v2f b0 = *(const v2f*)(&Ys[l15 * 256 + kk]);
        v2f b1 = *(const v2f*)(&Ys[(16 + l15) * 256 + kk]);
        c0 = wmma_k4(av, b0, c0);
        c1 = wmma_k4(av, b1, c1);
      }
#pragma unroll
      for (int j = 0; j < 8; ++j) {
        // exclude padded column 31 (N-tile 1, lane l15==15)
        float mv = fmaxf(c0[j], (l15 == 15) ? -1e30f : c1[j]);
        mv = fmaxf(mv, __shfl_xor(mv, 8, 32));   // stays within 16-lane halves
        mv = fmaxf(mv, __shfl_xor(mv, 4, 32));
        mv = fmaxf(mv, __shfl_xor(mv, 2, 32));
        mv = fmaxf(mv, __shfl_xor(mv, 1, 32));
        if (l15 == 0) {
          const int row = mt * 16 + j + 8 * half;   // lanes 0 & 16 write
          uniq[row] = fminf(uniq[row], 0.5f * (1.0f - mv));
        }
      }
    }
    __syncthreads();
  }

  // Final: logits = [pos, sim_neg(124)]; loss_b = logsumexp - pos.
  float* logits = Ys;   // reuse LDS
  if (t < 124) {
    const int n = t >> 2, s = t & 3;
    const int a = n + (n >= b);
    logits[1 + t] = lsort[(b * 32 + a) * 64 + sel_all[s * 31 + n]];
  }
  if (t == 0) logits[0] = lsort[(b * 32 + b) * 64];
  __syncthreads();
  if (t == 0) {
    float mx = logits[0];
    for (int i = 1; i < 125; ++i) mx = fmaxf(mx, logits[i]);
    float ssum = 0.0f;
    for (int i = 0; i < 125; ++i) ssum += expf(logits[i] - mx);
    lossb[b] = mx + logf(ssum) - logits[0];
  }
}

// ---------------------------------------------------------------------------
// Kernel 5: mean of 32 per-batch losses.
// ---------------------------------------------------------------------------
__global__ __launch_bounds__(32) void k_mean(const float* __restrict__ lossb,
                                             float* __restrict__ out) {
  if (threadIdx.x == 0) {
    float s = 0.0f;
    for (int i = 0; i < 32; ++i) s += lossb[i];
    out[0] = s / 32.0f;
  }
}

extern "C" void kernel_launch(void* const* d_in, const int* in_sizes, int n_in,
                              void* d_out, int out_size, void* d_ws, size_t ws_size,
                              hipStream_t stream) {
  const float* vis = (const float*)d_in[0];   // (32, 64, 256) f32
  const float* lan = (const float*)d_in[1];   // (32, 1, 256)  f32

  char* ws     = (char*)d_ws;
  float* sim   = (float*)(ws + 0);                       // 65536 f32 (256 KB)
  float* lsort = (float*)(ws + 262144);                  // 65536 f32
  int*   sidx  = (int*)  (ws + 524288);                  // 65536 i32
  float* vnorm = (float*)(ws + 786432);                  // 524288 f32 (2 MB)
  float* lossb = (float*)(ws + 786432 + 2097152);        // 32 f32
  float* out   = (float*)d_out;

  k_sim <<<dim3(128, 2), 32, 0, stream>>>(vis, lan, sim);
  k_sort<<<1024, 64, 0, stream>>>(sim, lsort, sidx);
  k_norm<<<2048, 256, 0, stream>>>(vis, vnorm);
  k_sel <<<32, 256, 0, stream>>>(vnorm, lsort, sidx, lossb);
  k_mean<<<1, 32, 0, stream>>>(lossb, out);
}